// HAN_Agg_9560597201181
// MI455X (gfx1250) — compile-verified
//
#include <hip/hip_runtime.h>
#include <math.h>

#define DIM 256
#define HEADS 8
#define REL 4
#define GAT_SLOPE 0.2f
#define SEM_SLOPE 0.01f

typedef __attribute__((ext_vector_type(2))) float v2f;
typedef __attribute__((ext_vector_type(8))) float v8f;

__device__ __forceinline__ float lrelu(float v, float s) { return v > 0.f ? v : s * v; }
// monotonic float<->int mapping for atomicMax on floats
__device__ __forceinline__ int fmap(float f) {
    int i = __float_as_int(f);
    return i < 0 ? (i ^ 0x7fffffff) : i;
}
__device__ __forceinline__ float funmap(int i) {
    return __int_as_float(i < 0 ? (i ^ 0x7fffffff) : i);
}

// ---------------------------------------------------------------------------
// WMMA GEMM: hs[r] = x_src[r] @ W_src[r], fp32, 16x16x4 WMMA.
// Block = 4 waves (128 threads) -> 16 rows x 64 cols. LDS staged.
// ---------------------------------------------------------------------------
__global__ void __launch_bounds__(128) gemm_hs_kernel(
    const float* __restrict__ X, const float* __restrict__ W,
    float* __restrict__ HS, int N)
{
    const int r    = blockIdx.z;
    const int m0   = blockIdx.x << 4;
    const int nblk = blockIdx.y << 6;
    const float* A = X + (size_t)r * N * DIM;
    const float* B = W + (size_t)r * DIM * DIM;
    float* C       = HS + (size_t)r * N * DIM;

    __shared__ float sA[16 * 68];   // pad stride 68 to avoid bank conflicts
    __shared__ float sB[64 * 64];

    const int tid  = threadIdx.x;
    const int wv   = tid >> 5;
    const int lane = tid & 31;
    const int hf   = lane >> 4;     // lane-half selects K pair / M half
    const int lr   = lane & 15;
    const int n0   = wv << 4;

    v8f acc = {};
    for (int k0 = 0; k0 < DIM; k0 += 64) {
        for (int idx = tid; idx < 16 * 64; idx += 128) {
            int row = idx >> 6, kk = idx & 63;
            sA[row * 68 + kk] = A[(size_t)(m0 + row) * DIM + k0 + kk];
        }
        for (int idx = tid; idx < 64 * 64; idx += 128) {
            int kk = idx >> 6, col = idx & 63;
            sB[idx] = B[(size_t)(k0 + kk) * DIM + nblk + col];
        }
        __syncthreads();
#pragma unroll
        for (int kk = 0; kk < 64; kk += 4) {
            v2f a, b;
            a.x = sA[lr * 68 + kk + 2 * hf + 0];
            a.y = sA[lr * 68 + kk + 2 * hf + 1];
            b.x = sB[(kk + 2 * hf + 0) * 64 + n0 + lr];
            b.y = sB[(kk + 2 * hf + 1) * 64 + n0 + lr];
            acc = __builtin_amdgcn_wmma_f32_16x16x4_f32(
                false, a, false, b, (short)0, acc, false, false);
        }
        __syncthreads();
    }
#pragma unroll
    for (int j = 0; j < 8; ++j)
        C[(size_t)(m0 + hf * 8 + j) * DIM + nblk + n0 + lr] = acc[j];
}

// ---------------------------------------------------------------------------
// Fold attention vectors into the weight matrices:
// wf[r][k][h] = sum_c W[r][k][h*32+c] * att[r][h][c]
// ---------------------------------------------------------------------------
__global__ void fold_kernel(const float* __restrict__ Ws, const float* __restrict__ Wd,
                            const float* __restrict__ As, const float* __restrict__ Ad,
                            float* __restrict__ WFs, float* __restrict__ WFd)
{
    int idx = blockIdx.x * blockDim.x + threadIdx.x;   // ((r*256+k)*8+h)
    if (idx >= REL * DIM * HEADS) return;
    int h = idx & 7;
    int k = (idx >> 3) & 255;
    int r = idx >> 11;
    float ss = 0.f, sd = 0.f;
#pragma unroll
    for (int c = 0; c < 32; ++c) {
        ss += Ws[((size_t)r * DIM + k) * DIM + h * 32 + c] * As[(r * HEADS + h) * 32 + c];
        sd += Wd[((size_t)r * DIM + k) * DIM + h * 32 + c] * Ad[(r * HEADS + h) * 32 + c];
    }
    WFs[idx] = ss;
    WFd[idx] = sd;
}

// ---------------------------------------------------------------------------
// al_s[r][n][h] = x_src[r][n] . wf_s[r][:,h] ; al_d from x_node/wf_d.
// Also self-loop alpha + monotonic-int max init.
// ---------------------------------------------------------------------------
__global__ void al_kernel(const float* __restrict__ Xs, const float* __restrict__ Xn,
                          const float* __restrict__ WFs, const float* __restrict__ WFd,
                          float* __restrict__ AlS, float* __restrict__ AlD,
                          float* __restrict__ SelfA, int* __restrict__ Mm, int N)
{
    int idx = blockIdx.x * blockDim.x + threadIdx.x;   // (r*N+n)*8+h
    if (idx >= REL * N * HEADS) return;
    int h  = idx & 7;
    int rn = idx >> 3;
    int r  = rn / N;
    const float* xs  = Xs + (size_t)rn * DIM;
    const float* xn  = Xn + (size_t)(rn - r * N) * DIM;
    const float* wfs = WFs + r * DIM * HEADS;
    const float* wfd = WFd + r * DIM * HEADS;
    float ss = 0.f, sd = 0.f;
    for (int k = 0; k < DIM; ++k) {
        ss += xs[k] * wfs[k * HEADS + h];
        sd += xn[k] * wfd[k * HEADS + h];
    }
    AlS[idx] = ss;
    AlD[idx] = sd;
    float sa = lrelu(ss + sd, GAT_SLOPE);
    SelfA[idx] = sa;
    Mm[idx] = fmap(sa);   // self loop participates in segment max
}

// edges layout [4][2][E]; reference flips: src = edges[r][1], dst = edges[r][0]
__global__ void edge_max_kernel(const int* __restrict__ Ed,
                                const float* __restrict__ AlS, const float* __restrict__ AlD,
                                int* __restrict__ Mm, int N, int E)
{
    long long idx = (long long)blockIdx.x * blockDim.x + threadIdx.x;
    if (idx >= (long long)REL * E * HEADS) return;
    int h = (int)(idx & 7);
    long long re = idx >> 3;
    int e = (int)(re % E);
    int r = (int)(re / E);
    int src = Ed[((size_t)r * 2 + 1) * E + e];
    int dst = Ed[((size_t)r * 2 + 0) * E + e];
    // masked (src==dst) edges still contribute to the max, per reference
    float alpha = lrelu(AlS[((size_t)r * N + src) * HEADS + h] +
                        AlD[((size_t)r * N + dst) * HEADS + h], GAT_SLOPE);
    atomicMax(&Mm[((size_t)r * N + dst) * HEADS + h], fmap(alpha));
}

__global__ void den_init_kernel(const float* __restrict__ SelfA, const int* __restrict__ Mm,
                                float* __restrict__ Den, int N)
{
    int idx = blockIdx.x * blockDim.x + threadIdx.x;
    if (idx >= REL * N * HEADS) return;
    Den[idx] = expf(SelfA[idx] - funmap(Mm[idx]));   // self-loop term
}

__global__ void edge_den_kernel(const int* __restrict__ Ed,
                                const float* __restrict__ AlS, const float* __restrict__ AlD,
                                const int* __restrict__ Mm, float* __restrict__ Den,
                                int N, int E)
{
    long long idx = (long long)blockIdx.x * blockDim.x + threadIdx.x;
    if (idx >= (long long)REL * E * HEADS) return;
    int h = (int)(idx & 7);
    long long re = idx >> 3;
    int e = (int)(re % E);
    int r = (int)(re / E);
    int src = Ed[((size_t)r * 2 + 1) * E + e];
    int dst = Ed[((size_t)r * 2 + 0) * E + e];
    if (src == dst) return;   // masked out of exp-sum
    float alpha = lrelu(AlS[((size_t)r * N + src) * HEADS + h] +
                        AlD[((size_t)r * N + dst) * HEADS + h], GAT_SLOPE);
    int di = ((size_t)r * N + dst) * HEADS + h;
    atomicAdd(&Den[di], expf(alpha - funmap(Mm[di])));
}

__global__ void agg_init_kernel(const float* __restrict__ HS, const float* __restrict__ SelfA,
                                const int* __restrict__ Mm, const float* __restrict__ Den,
                                float* __restrict__ Agg, int N)
{
    int idx = blockIdx.x * blockDim.x + threadIdx.x;   // (r*N+n)*256+d
    if (idx >= REL * N * DIM) return;
    int d  = idx & 255;
    int rn = idx >> 8;
    int ai = rn * HEADS + (d >> 5);
    float w = expf(SelfA[ai] - funmap(Mm[ai])) / Den[ai];
    Agg[idx] = w * HS[idx];
}

// heavy pass: one 256-thread block per edge, gather hs[src], atomic scatter agg[dst]
__global__ void __launch_bounds__(256) edge_agg_kernel(
    const int* __restrict__ Ed, const float* __restrict__ HS,
    const float* __restrict__ AlS, const float* __restrict__ AlD,
    const int* __restrict__ Mm, const float* __restrict__ Den,
    float* __restrict__ Agg, int N, int E)
{
    int be = blockIdx.x;          // r*E + e
    int r = be / E, e = be % E;
    int src = Ed[((size_t)r * 2 + 1) * E + e];
    int dst = Ed[((size_t)r * 2 + 0) * E + e];
    if (src == dst) return;       // masked edge
    const float* hsrow = HS + ((size_t)r * N + src) * DIM;
    __builtin_prefetch(hsrow + threadIdx.x, 0, 0);   // global_prefetch_b8
    int d = threadIdx.x;
    int h = d >> 5;
    int di = ((size_t)r * N + dst) * HEADS + h;
    float alpha = lrelu(AlS[((size_t)r * N + src) * HEADS + h] + AlD[di], GAT_SLOPE);
    float w = expf(alpha - funmap(Mm[di])) / Den[di];
    atomicAdd(&Agg[((size_t)r * N + dst) * DIM + d], w * hsrow[d]);
}

// ---------------------------------------------------------------------------
// Semantic attention, fused per node: relu(agg+bias), dot with u, softmax
// over relations, weighted combine.
// ---------------------------------------------------------------------------
__global__ void __launch_bounds__(256) semantic_kernel(
    const float* __restrict__ Agg, const float* __restrict__ Xn,
    const float* __restrict__ Bias, const float* __restrict__ U,
    float* __restrict__ Comb, int N)
{
    const int n = blockIdx.x;
    const int d = threadIdx.x;
    const int wv = d >> 5, lane = d & 31;
    float x = Xn[(size_t)n * DIM + d];
    float a[REL];
#pragma unroll
    for (int r = 0; r < REL; ++r) {
        float t = Agg[((size_t)r * N + n) * DIM + d] + Bias[r * DIM + d];
        a[r] = t > 0.f ? t : 0.f;   // HAN_NeighAgg relu
    }
    float p[5];
#pragma unroll
    for (int r = 0; r < REL; ++r) p[r] = a[r] * U[d];
    p[4] = x * U[DIM + d];

    __shared__ float red[5][8];
#pragma unroll
    for (int i = 0; i < 5; ++i) {
        float s = p[i];
#pragma unroll
        for (int m = 16; m > 0; m >>= 1) s += __shfl_xor(s, m, 32);
        if (lane == 0) red[i][wv] = s;
    }
    __syncthreads();
    float dot[5];
#pragma unroll
    for (int i = 0; i < 5; ++i) {
        float s = 0.f;
#pragma unroll
        for (int w = 0; w < 8; ++w) s += red[i][w];
        dot[i] = s;
    }
    float sc[REL], ssum = 0.f;
#pragma unroll
    for (int r = 0; r < REL; ++r) {
        sc[r] = expf(lrelu(dot[r] + dot[4], SEM_SLOPE));
        ssum += sc[r];
    }
    float c = 0.f;
#pragma unroll
    for (int r = 0; r < REL; ++r) c += (sc[r] / ssum) * a[r];
    Comb[(size_t)n * DIM + d] = c;
}

// ---------------------------------------------------------------------------
// Final WMMA GEMM: relu([x_node | combined] @ lin_W + lin_b)  (K = 512)
// ---------------------------------------------------------------------------
__global__ void __launch_bounds__(128) gemm_final_kernel(
    const float* __restrict__ Xn, const float* __restrict__ Comb,
    const float* __restrict__ Wl, const float* __restrict__ Bl,
    float* __restrict__ Out, int N)
{
    const int m0   = blockIdx.x << 4;
    const int nblk = blockIdx.y << 6;
    __shared__ float sA[16 * 68];
    __shared__ float sB[64 * 64];
    const int tid  = threadIdx.x;
    const int wv   = tid >> 5;
    const int lane = tid & 31;
    const int hf   = lane >> 4;
    const int lr   = lane & 15;
    const int n0   = wv << 4;

    v8f acc = {};
    for (int k0 = 0; k0 < 2 * DIM; k0 += 64) {
        const float* Asrc = (k0 < DIM) ? Xn : Comb;   // virtual concat
        int ka = k0 & (DIM - 1);
        for (int idx = tid; idx < 16 * 64; idx += 128) {
            int row = idx >> 6, kk = idx & 63;
            sA[row * 68 + kk] = Asrc[(size_t)(m0 + row) * DIM + ka + kk];
        }
        for (int idx = tid; idx < 64 * 64; idx += 128) {
            int kk = idx >> 6, col = idx & 63;
            sB[idx] = Wl[(size_t)(k0 + kk) * DIM + nblk + col];
        }
        __syncthreads();
#pragma unroll
        for (int kk = 0; kk < 64; kk += 4) {
            v2f a, b;
            a.x = sA[lr * 68 + kk + 2 * hf + 0];
            a.y = sA[lr * 68 + kk + 2 * hf + 1];
            b.x = sB[(kk + 2 * hf + 0) * 64 + n0 + lr];
            b.y = sB[(kk + 2 * hf + 1) * 64 + n0 + lr];
            acc = __builtin_amdgcn_wmma_f32_16x16x4_f32(
                false, a, false, b, (short)0, acc, false, false);
        }
        __syncthreads();
    }
#pragma unroll
    for (int j = 0; j < 8; ++j) {
        int col = nblk + n0 + lr;
        float v = acc[j] + Bl[col];
        Out[(size_t)(m0 + hf * 8 + j) * DIM + col] = v > 0.f ? v : 0.f;
    }
}

// in-place row L2 normalize, one wave per row
__global__ void __launch_bounds__(256) normalize_kernel(float* __restrict__ Out, int N)
{
    int row  = blockIdx.x * 8 + (threadIdx.x >> 5);
    int lane = threadIdx.x & 31;
    if (row >= N) return;
    float v[8];
    float ss = 0.f;
#pragma unroll
    for (int j = 0; j < 8; ++j) {
        v[j] = Out[(size_t)row * DIM + lane + 32 * j];
        ss += v[j] * v[j];
    }
#pragma unroll
    for (int m = 16; m > 0; m >>= 1) ss += __shfl_xor(ss, m, 32);
    float nrm = sqrtf(ss);
    nrm = nrm > 1e-12f ? nrm : 1e-12f;
    float inv = 1.0f / nrm;
#pragma unroll
    for (int j = 0; j < 8; ++j)
        Out[(size_t)row * DIM + lane + 32 * j] = v[j] * inv;
}

// ---------------------------------------------------------------------------
extern "C" void kernel_launch(void* const* d_in, const int* in_sizes, int n_in,
                              void* d_out, int out_size, void* d_ws, size_t ws_size,
                              hipStream_t stream)
{
    const float* x_src  = (const float*)d_in[0];
    const float* x_node = (const float*)d_in[1];
    const int*   edges  = (const int*)d_in[2];
    // d_in[3] = ew: unused by reference (GATConv has no edge features)
    const float* W_src  = (const float*)d_in[4];
    const float* W_dst  = (const float*)d_in[5];
    const float* att_s  = (const float*)d_in[6];
    const float* att_d  = (const float*)d_in[7];
    const float* bias   = (const float*)d_in[8];
    const float* u      = (const float*)d_in[9];
    const float* lin_W  = (const float*)d_in[10];
    const float* lin_b  = (const float*)d_in[11];

    const int N = in_sizes[1] / DIM;          // 20000
    const int E = in_sizes[2] / (2 * REL);    // 320000

    char* ws = (char*)d_ws;
    auto carve = [&](size_t bytes) -> char* {
        char* p = ws;
        ws += (bytes + 255) & ~(size_t)255;
        return p;
    };
    float* HS    = (float*)carve((size_t)REL * N * DIM * sizeof(float));
    float* Agg   = (float*)carve((size_t)REL * N * DIM * sizeof(float));
    float* Comb  = (float*)carve((size_t)N * DIM * sizeof(float));
    float* AlS   = (float*)carve((size_t)REL * N * HEADS * sizeof(float));
    float* AlD   = (float*)carve((size_t)REL * N * HEADS * sizeof(float));
    float* SelfA = (float*)carve((size_t)REL * N * HEADS * sizeof(float));
    int*   Mm    = (int*)  carve((size_t)REL * N * HEADS * sizeof(int));
    float* Den   = (float*)carve((size_t)REL * N * HEADS * sizeof(float));
    float* WFs   = (float*)carve((size_t)REL * DIM * HEADS * sizeof(float));
    float* WFd   = (float*)carve((size_t)REL * DIM * HEADS * sizeof(float));
    (void)ws_size; (void)n_in; (void)out_size;

    // 1. hs[r] = x_src[r] @ W_src[r]  (WMMA)
    gemm_hs_kernel<<<dim3(N / 16, DIM / 64, REL), 128, 0, stream>>>(x_src, W_src, HS, N);

    // 2. fold att into W, compute al_s/al_d + self-loop alpha + max init
    fold_kernel<<<(REL * DIM * HEADS + 255) / 256, 256, 0, stream>>>(
        W_src, W_dst, att_s, att_d, WFs, WFd);
    al_kernel<<<(REL * N * HEADS + 255) / 256, 256, 0, stream>>>(
        x_src, x_node, WFs, WFd, AlS, AlD, SelfA, Mm, N);

    // 3. segment softmax + weighted aggregation over edges
    long long nEdgeT = (long long)REL * E * HEADS;
    int edgeBlocks = (int)((nEdgeT + 255) / 256);
    edge_max_kernel<<<edgeBlocks, 256, 0, stream>>>(edges, AlS, AlD, Mm, N, E);
    den_init_kernel<<<(REL * N * HEADS + 255) / 256, 256, 0, stream>>>(SelfA, Mm, Den, N);
    edge_den_kernel<<<edgeBlocks, 256, 0, stream>>>(edges, AlS, AlD, Mm, Den, N, E);
    agg_init_kernel<<<(REL * N * DIM + 255) / 256, 256, 0, stream>>>(HS, SelfA, Mm, Den, Agg, N);
    edge_agg_kernel<<<REL * E, 256, 0, stream>>>(edges, HS, AlS, AlD, Mm, Den, Agg, N, E);

    // 4. semantic attention -> combined
    semantic_kernel<<<N, 256, 0, stream>>>(Agg, x_node, bias, u, Comb, N);

    // 5. final GEMM (WMMA, K=512 virtual concat) + relu, then row normalize
    gemm_final_kernel<<<dim3(N / 16, DIM / 64), 128, 0, stream>>>(
        x_node, Comb, lin_W, lin_b, (float*)d_out, N);
    normalize_kernel<<<(N + 7) / 8, 256, 0, stream>>>((float*)d_out, N);
}